// LogicalOperatorFusion_10711648436289
// MI455X (gfx1250) — compile-verified
//
#include <hip/hip_runtime.h>
#include <math.h>

typedef __attribute__((ext_vector_type(2))) float v2f;
typedef __attribute__((ext_vector_type(8))) float v8f;

#define HDIM 512
#define LDS_STRIDE 516  // 512 + 4 pad: conflict-free column reads across 16 lanes

__device__ __forceinline__ v8f wmma_f32_k4(v2f a, v2f b, v8f c) {
  // D = A(16x4,f32) * B(4x16,f32) + C(16x16,f32)
  return __builtin_amdgcn_wmma_f32_16x16x4_f32(
      false, a, false, b, (short)0, c, false, false);
}

// One wave: 32(M) x 64(N) strip, K step 4, using TRANSPOSED weights Wt[N][K].
// B fragment = single b64 (W[k+2h][n], W[k+2h+1][n] contiguous in Wt).
// B frags shared across both M tiles: 6 loads per 8 WMMAs.
template <int K>
__device__ __forceinline__ void gemm_strip2(const float* __restrict__ x0,
                                            const float* __restrict__ x1,
                                            const float* __restrict__ Wt,
                                            int nb, int half,
                                            v8f acc0[4], v8f acc1[4]) {
#pragma unroll 2
  for (int k0 = 0; k0 < K; k0 += 4) {
    const int kk = k0 + 2 * half;
    v2f a0 = *(const v2f*)(x0 + kk);
    v2f a1 = *(const v2f*)(x1 + kk);
#pragma unroll
    for (int t = 0; t < 4; ++t) {
      int n = nb + t * 16;
      v2f b = *(const v2f*)(Wt + (size_t)n * K + kk);
      acc0[t] = wmma_f32_k4(a0, b, acc0[t]);
      acc1[t] = wmma_f32_k4(a1, b, acc1[t]);
    }
  }
}

// ---- weight transpose (one-time, ~7 MB into d_ws) ----
__global__ __launch_bounds__(256) void transpose_kernel(
    const float* __restrict__ in, float* __restrict__ outT, int K, int N) {
  __shared__ float tile[32][33];
  const int k0 = blockIdx.x * 32, n0 = blockIdx.y * 32;
  const int tx = threadIdx.x & 31, ty = threadIdx.x >> 5;  // 32 x 8
#pragma unroll
  for (int r = 0; r < 32; r += 8)
    tile[ty + r][tx] = in[(size_t)(k0 + ty + r) * N + n0 + tx];
  __syncthreads();
#pragma unroll
  for (int r = 0; r < 32; r += 8)
    outT[(size_t)(n0 + ty + r) * K + k0 + tx] = tile[tx][ty + r];
}

// Fused Linear(2H->H) -> ReLU -> Linear(H->H); block = 32 rows x 512 cols.
__global__ __launch_bounds__(256) void fused_and_or_kernel(
    const float* __restrict__ fv,
    const float* __restrict__ W1t, const float* __restrict__ b1,
    const float* __restrict__ W2t, const float* __restrict__ b2,
    float* __restrict__ out, int is_or) {
  __shared__ float hbuf[32 * LDS_STRIDE];

  const int tid  = threadIdx.x;
  const int wave = tid >> 5;
  const int lane = tid & 31;
  const int half = lane >> 4;
  const int lrow = lane & 15;
  const int nb   = wave * 64 + lrow;
  const int m0   = blockIdx.x * 32;
  const int ma   = m0 + lrow;        // M tile 0 row for this lane
  const int mb   = m0 + 16 + lrow;   // M tile 1 row

  // Pair rows are contiguous 1024-float slices: fv[b, 2p] ++ fv[b, 2p+1]
  auto arow = [&](int m) -> size_t {
    return is_or ? (size_t)(m >> 3) * 32768u + 16384u + (size_t)(m & 7) * 1024u
                 : (size_t)(m >> 4) * 32768u + (size_t)(m & 15) * 1024u;
  };

  v8f a0[4] = {}, a1[4] = {};
  gemm_strip2<1024>(fv + arow(ma), fv + arow(mb), W1t, nb, half, a0, a1);

  // h = relu(acc + b1) -> LDS
#pragma unroll
  for (int t = 0; t < 4; ++t) {
    int n = nb + t * 16;
    float bv = b1[n];
#pragma unroll
    for (int r = 0; r < 8; ++r) {
      hbuf[(r + 8 * half) * LDS_STRIDE + n]      = fmaxf(a0[t][r] + bv, 0.0f);
      hbuf[(16 + r + 8 * half) * LDS_STRIDE + n] = fmaxf(a1[t][r] + bv, 0.0f);
    }
  }
  __syncthreads();

  v8f c0[4] = {}, c1[4] = {};
  gemm_strip2<512>(&hbuf[lrow * LDS_STRIDE], &hbuf[(16 + lrow) * LDS_STRIDE],
                   W2t, nb, half, c0, c1);

  // y = acc + b2 -> d_out[B, 40, 512]; and -> ch 0..15, or -> ch 16..23
  auto yrow = [&](int m) -> size_t {
    return is_or ? ((size_t)(m >> 3) * 40u + 16u + (size_t)(m & 7)) * 512u
                 : ((size_t)(m >> 4) * 40u + (size_t)(m & 15)) * 512u;
  };
#pragma unroll
  for (int t = 0; t < 4; ++t) {
    int n = nb + t * 16;
    float bv = b2[n];
#pragma unroll
    for (int r = 0; r < 8; ++r) {
      int mm = m0 + r + 8 * half;
      out[yrow(mm) + n]      = c0[t][r] + bv;
      out[yrow(mm + 16) + n] = c1[t][r] + bv;
    }
  }
}

// not: tanh(fv[:, 48+j] @ not_W + not_b) -> output channels 24..31
__global__ __launch_bounds__(256) void not_tanh_kernel(
    const float* __restrict__ fv, const float* __restrict__ Wt,
    const float* __restrict__ b, float* __restrict__ out) {
  const int tid  = threadIdx.x;
  const int wave = tid >> 5;
  const int lane = tid & 31;
  const int half = lane >> 4;
  const int lrow = lane & 15;
  const int nb   = wave * 64 + lrow;
  const int m0   = blockIdx.x * 32;

  auto arow = [&](int m) -> size_t {
    return (size_t)(m >> 3) * 32768u + (size_t)(48 + (m & 7)) * 512u;
  };

  v8f c0[4] = {}, c1[4] = {};
  gemm_strip2<512>(fv + arow(m0 + lrow), fv + arow(m0 + 16 + lrow),
                   Wt, nb, half, c0, c1);

  auto yrow = [&](int m) -> size_t {
    return ((size_t)(m >> 3) * 40u + 24u + (size_t)(m & 7)) * 512u;
  };
#pragma unroll
  for (int t = 0; t < 4; ++t) {
    int n = nb + t * 16;
    float bv = b[n];
#pragma unroll
    for (int r = 0; r < 8; ++r) {
      int mm = m0 + r + 8 * half;
      out[yrow(mm) + n]      = tanhf(c0[t][r] + bv);
      out[yrow(mm + 16) + n] = tanhf(c1[t][r] + bv);
    }
  }
}

// fields 56..63 (untouched) -> output channels 32..39; float4, coalesced
__global__ __launch_bounds__(256) void copy_leftover_kernel(
    const float* __restrict__ fv, float* __restrict__ out) {
  size_t i = (size_t)blockIdx.x * blockDim.x + threadIdx.x;  // float4 index
  size_t e = i * 4;                                          // elem in [B,8,512]
  size_t bidx = e / (8 * 512);
  size_t rem  = e % (8 * 512);
  size_t j = rem / 512, h = rem % 512;
  float4 v = *(const float4*)(fv + ((bidx * 64 + 56 + j) * 512 + h));
  *(float4*)(out + ((bidx * 40 + 32 + j) * 512 + h)) = v;
}

extern "C" void kernel_launch(void* const* d_in, const int* in_sizes, int n_in,
                              void* d_out, int out_size, void* d_ws, size_t ws_size,
                              hipStream_t stream) {
  const float* fv     = (const float*)d_in[0];
  const float* and_W1 = (const float*)d_in[1];
  const float* and_b1 = (const float*)d_in[2];
  const float* and_W2 = (const float*)d_in[3];
  const float* and_b2 = (const float*)d_in[4];
  const float* or_W1  = (const float*)d_in[5];
  const float* or_b1  = (const float*)d_in[6];
  const float* or_W2  = (const float*)d_in[7];
  const float* or_b2  = (const float*)d_in[8];
  const float* not_W  = (const float*)d_in[9];
  const float* not_b  = (const float*)d_in[10];
  float* out = (float*)d_out;
  float* ws  = (float*)d_ws;

  // Transposed weights in workspace (float offsets)
  float* aW1t = ws;                    // 512*1024
  float* aW2t = ws + 524288;           // 512*512
  float* oW1t = ws + 786432;           // 512*1024
  float* oW2t = ws + 1310720;          // 512*512
  float* nWt  = ws + 1572864;          // 512*512  (total 7.0 MB)

  dim3 g2h(32, 16), gh(16, 16);  // (K/32, N/32)
  transpose_kernel<<<g2h, 256, 0, stream>>>(and_W1, aW1t, 1024, 512);
  transpose_kernel<<<gh,  256, 0, stream>>>(and_W2, aW2t, 512, 512);
  transpose_kernel<<<g2h, 256, 0, stream>>>(or_W1,  oW1t, 1024, 512);
  transpose_kernel<<<gh,  256, 0, stream>>>(or_W2,  oW2t, 512, 512);
  transpose_kernel<<<gh,  256, 0, stream>>>(not_W,  nWt,  512, 512);

  // and: 16384 rows / 32 = 512 blocks; or/not: 8192 rows / 32 = 256 blocks
  fused_and_or_kernel<<<512, 256, 0, stream>>>(fv, aW1t, and_b1, aW2t, and_b2,
                                               out, 0);
  fused_and_or_kernel<<<256, 256, 0, stream>>>(fv, oW1t, or_b1, oW2t, or_b2,
                                               out, 1);
  not_tanh_kernel<<<256, 256, 0, stream>>>(fv, nWt, not_b, out);
  copy_leftover_kernel<<<4096, 256, 0, stream>>>(fv, out);
}